// STGNN_45440753992335
// MI455X (gfx1250) — compile-verified
//
#include <hip/hip_runtime.h>

typedef __attribute__((ext_vector_type(2))) float v2f;
typedef __attribute__((ext_vector_type(8))) float v8f;

#define N_NODES 50000
#define N_EDGES 800000
#define D       96
#define DOUT    4

struct F3 { float x, y, z; };   // 12-byte, 4-aligned -> global_load_b96 eligible

// ---------------- degree / norm ----------------
__global__ __launch_bounds__(256) void deg_init(float* __restrict__ deg) {
    int i = blockIdx.x * blockDim.x + threadIdx.x;
    if (i < N_NODES) deg[i] = 1.0f;            // self-loop
}

__global__ __launch_bounds__(256) void deg_count(const int* __restrict__ dst,
                                                 float* __restrict__ deg) {
    int e = blockIdx.x * blockDim.x + threadIdx.x;
    if (e < N_EDGES) atomicAdd(&deg[dst[e]], 1.0f);
}

__global__ __launch_bounds__(256) void deg_rsqrt(float* __restrict__ deg) {
    int i = blockIdx.x * blockDim.x + threadIdx.x;
    if (i < N_NODES) deg[i] = rsqrtf(deg[i]);  // deg >= 1 always (self-loop)
}

// ---------------- hp = dinv * (X @ W); agg = hp (self-loop term) ------------
// grid.x = 3125 M-tiles; 6 waves/block = the 6 N-tiles; W staged in LDS.
__global__ __launch_bounds__(192) void gemm_scale(const float* __restrict__ X,
                                                  const float* __restrict__ W,
                                                  const float* __restrict__ dinv,
                                                  float* __restrict__ hp,
                                                  float* __restrict__ agg) {
    __shared__ float Ws[D * D];                // 36 KB of the 320 KB WGP LDS
    const int tid = threadIdx.x;

    // cooperative 16B-vector load of W into LDS: 9216 floats / 192 threads
    {
        const float4* Wv  = (const float4*)W;
        float4*       Wsv = (float4*)Ws;
#pragma unroll
        for (int i = 0; i < 12; ++i) Wsv[tid + i * 192] = Wv[tid + i * 192];
    }
    __syncthreads();

    const int wave = tid >> 5;                 // 0..5 -> N tile
    const int lane = tid & 31;
    const int g    = lane >> 4;                // lane group (K split)
    const int r16  = lane & 15;                // M for A-frag, N for B/C-frag
    const int m0   = blockIdx.x * 16;
    const int n0   = wave * 16;

    const float* Arow = X + (size_t)(m0 + r16) * D;

    v8f c = {};
#pragma unroll
    for (int kk = 0; kk < D; kk += 4) {
        const int ka = kk + g * 2;             // even -> 8B aligned
        // A 16x4 fp32 frag: lane holds (M=r16, K=ka..ka+1)
        v2f a = *(const v2f*)(Arow + ka);
        // B 4x16 fp32 frag: vgpr j holds (K = 2g+j, N = n0+r16)
        v2f b;
        b.x = Ws[(ka + 0) * D + n0 + r16];
        b.y = Ws[(ka + 1) * D + n0 + r16];
        c = __builtin_amdgcn_wmma_f32_16x16x4_f32(false, a, false, b,
                                                  (short)0, c, false, false);
    }

    // C/D layout: vgpr r -> row m0 + r + 8*g, col n0 + r16
#pragma unroll
    for (int r = 0; r < 8; ++r) {
        const int row   = m0 + r + 8 * g;
        const float v   = c[r] * dinv[row];
        const size_t ix = (size_t)row * D + n0 + r16;
        hp[ix]  = v;
        agg[ix] = v;   // self-loop contribution; also zero-initializes agg
    }
}

// ---------------- edge scatter: agg[dst] += hp[src] -------------------------
// One wave per edge; lane handles 3 contiguous features (96 = 32*3).
// Gather = contiguous 384B of hp[src]; atomics = contiguous 384B of agg[dst].
__global__ __launch_bounds__(256) void scatter_add(const int* __restrict__ src,
                                                   const int* __restrict__ dst,
                                                   const float* __restrict__ hp,
                                                   float* __restrict__ agg) {
    const int e = blockIdx.x * 8 + (threadIdx.x >> 5);
    if (e >= N_EDGES) return;
    const int lane = threadIdx.x & 31;
    const int s = src[e];                       // wave-uniform broadcast loads
    const int d = dst[e];
    const int f = lane * 3;
    const F3 v = *(const F3*)(hp + (size_t)s * D + f);
    float* a = agg + (size_t)d * D + f;
    atomicAdd(a + 0, v.x);
    atomicAdd(a + 1, v.y);
    atomicAdd(a + 2, v.z);
}

// ---------------- xout = relu(dinv * agg + b) -------------------------------
// One wave per node row; lane handles 3 contiguous features; 12B vector I/O.
__global__ __launch_bounds__(256) void finalize_relu(const float* __restrict__ agg,
                                                     const float* __restrict__ dinv,
                                                     const float* __restrict__ bias,
                                                     float* __restrict__ xout) {
    const int i = blockIdx.x * 8 + (threadIdx.x >> 5);
    if (i >= N_NODES) return;
    const int  f  = (threadIdx.x & 31) * 3;
    const float s = dinv[i];                    // wave-uniform
    const F3 a = *(const F3*)(agg + (size_t)i * D + f);
    const F3 b = *(const F3*)(bias + f);
    F3 o;
    o.x = fmaf(s, a.x, b.x); o.x = o.x > 0.0f ? o.x : 0.0f;
    o.y = fmaf(s, a.y, b.y); o.y = o.y > 0.0f ? o.y : 0.0f;
    o.z = fmaf(s, a.z, b.z); o.z = o.z > 0.0f ? o.z : 0.0f;
    *(F3*)(xout + (size_t)i * D + f) = o;
}

// ---------------- final FC: [N,96] @ [96,4] + b -----------------------------
__global__ __launch_bounds__(256) void fc_out(const float* __restrict__ h,
                                              const float* __restrict__ Wfc,
                                              const float* __restrict__ bfc,
                                              float* __restrict__ out) {
    const int tid = blockIdx.x * blockDim.x + threadIdx.x;
    if (tid >= N_NODES * DOUT) return;
    const int i = tid >> 2;
    const int o = tid & 3;
    const float* hr = h + (size_t)i * D;
    float acc = bfc[o];
#pragma unroll
    for (int f = 0; f < D; ++f) acc = fmaf(hr[f], Wfc[f * DOUT + o], acc);
    out[tid] = acc;
}

extern "C" void kernel_launch(void* const* d_in, const int* in_sizes, int n_in,
                              void* d_out, int out_size, void* d_ws, size_t ws_size,
                              hipStream_t stream) {
    const float* x   = (const float*)d_in[0];
    const int*   ei  = (const int*)d_in[1];     // [2, E]
    const int*   src = ei;
    const int*   dst = ei + N_EDGES;
    const float* W0  = (const float*)d_in[2];
    const float* b0  = (const float*)d_in[3];
    const float* W1  = (const float*)d_in[4];
    const float* b1  = (const float*)d_in[5];
    const float* W2  = (const float*)d_in[6];
    const float* b2  = (const float*)d_in[7];
    const float* Wfc = (const float*)d_in[8];
    const float* bfc = (const float*)d_in[9];

    char* ws = (char*)d_ws;
    const size_t BIG = (size_t)N_NODES * D * sizeof(float);   // 19.2 MB
    float* dinv = (float*)ws;                                 // 200 KB
    float* bufA = (float*)(ws + (1u << 18));
    float* bufB = (float*)(ws + (1u << 18) + BIG);
    float* agg  = (float*)(ws + (1u << 18) + 2 * BIG);

    const int gN  = (N_NODES + 255) / 256;
    const int gE  = (N_EDGES + 255) / 256;
    const int gEW = (N_EDGES + 7) / 8;          // 8 waves (edges) per block
    const int gNW = (N_NODES + 7) / 8;          // 8 waves (rows)  per block
    const int gO  = (N_NODES * DOUT + 255) / 256;

    // symmetric degree norm (shared by all 3 layers)
    deg_init <<<gN, 256, 0, stream>>>(dinv);
    deg_count<<<gE, 256, 0, stream>>>(dst, dinv);
    deg_rsqrt<<<gN, 256, 0, stream>>>(dinv);

    // layer 1: x -> bufA
    gemm_scale   <<<N_NODES / 16, 192, 0, stream>>>(x, W0, dinv, bufA, agg);
    scatter_add  <<<gEW, 256, 0, stream>>>(src, dst, bufA, agg);
    finalize_relu<<<gNW, 256, 0, stream>>>(agg, dinv, b0, bufA);

    // layer 2: bufA -> bufB
    gemm_scale   <<<N_NODES / 16, 192, 0, stream>>>(bufA, W1, dinv, bufB, agg);
    scatter_add  <<<gEW, 256, 0, stream>>>(src, dst, bufB, agg);
    finalize_relu<<<gNW, 256, 0, stream>>>(agg, dinv, b1, bufB);

    // layer 3: bufB -> bufA
    gemm_scale   <<<N_NODES / 16, 192, 0, stream>>>(bufB, W2, dinv, bufA, agg);
    scatter_add  <<<gEW, 256, 0, stream>>>(src, dst, bufA, agg);
    finalize_relu<<<gNW, 256, 0, stream>>>(agg, dinv, b2, bufA);

    // FC head
    fc_out<<<gO, 256, 0, stream>>>(bufA, Wfc, bfc, (float*)d_out);
}